// GATLayer_76836964925866
// MI455X (gfx1250) — compile-verified
//
#include <hip/hip_runtime.h>
#include <hip/hip_bf16.h>
#include <cstdint>

typedef __attribute__((ext_vector_type(2))) float v2f;
typedef __attribute__((ext_vector_type(8))) float v8f;

#define OUT_DIM 64
#define IN_DIM 256
#define NEG_SLOPE 0.01f

// ---------------- GEMM: z = h(100000x256) @ fc_w(256x64) via f32 WMMA -------
// Each wave: 16 rows x 64 cols. Block = 256 threads = 8 waves = 128 rows.
// fc_w (64 KB) staged in LDS via CDNA5 async global->LDS copies.
__global__ void k_gemm(const float* __restrict__ h, const float* __restrict__ fcw,
                       float* __restrict__ z, int n) {
    __shared__ float Bs[IN_DIM * OUT_DIM];   // 64 KB

    const int tid = threadIdx.x;
    {
        // 256 lanes x 16 iters x 16B = 64 KB, async DMA into LDS
        const char* g = (const char*)fcw;
        unsigned ldsBase = (unsigned)(uintptr_t)&Bs[0];
        size_t off0 = (size_t)tid * 16;
        #pragma unroll
        for (int it = 0; it < 16; ++it) {
            size_t byteOff = off0 + (size_t)it * 4096;
            uint64_t ga = (uint64_t)(g + byteOff);
            unsigned la = ldsBase + (unsigned)byteOff;
            asm volatile("global_load_async_to_lds_b128 %0, %1, off"
                         :: "v"(la), "v"(ga) : "memory");
        }
        asm volatile("s_wait_asynccnt 0x0" ::: "memory");
    }
    __syncthreads();

    const int lane = tid & 31;
    const int wave = tid >> 5;
    const int half = lane >> 4;   // 0: lanes 0-15, 1: lanes 16-31
    const int lm   = lane & 15;

    const long row0 = (long)blockIdx.x * 128 + (long)wave * 16;
    long ra = row0 + lm;
    if (ra > n - 1) ra = n - 1;               // clamp tail (EXEC must stay full for WMMA)
    const float* __restrict__ hrow = h + (size_t)ra * IN_DIM;

    v8f c0 = {}, c1 = {}, c2 = {}, c3 = {};

    for (int k0 = 0; k0 < IN_DIM; k0 += 4) {
        const int kk = k0 + half * 2;
        v2f a = *(const v2f*)(hrow + kk);                 // A 16x4 slice
        const float* bp = Bs + kk * OUT_DIM + lm;         // B 4x16 per N-tile
        v2f b0, b1, b2, b3;
        b0.x = bp[0];  b0.y = bp[OUT_DIM + 0];
        b1.x = bp[16]; b1.y = bp[OUT_DIM + 16];
        b2.x = bp[32]; b2.y = bp[OUT_DIM + 32];
        b3.x = bp[48]; b3.y = bp[OUT_DIM + 48];

        c0 = __builtin_amdgcn_wmma_f32_16x16x4_f32(false, a, false, b0, (short)0, c0, false, false);
        c1 = __builtin_amdgcn_wmma_f32_16x16x4_f32(false, a, false, b1, (short)0, c1, false, false);
        c2 = __builtin_amdgcn_wmma_f32_16x16x4_f32(false, a, false, b2, (short)0, c2, false, false);
        c3 = __builtin_amdgcn_wmma_f32_16x16x4_f32(false, a, false, b3, (short)0, c3, false, false);
    }

    for (int r = 0; r < 8; ++r) {
        long row = row0 + r + half * 8;
        if (row < n) {
            float* zr = z + (size_t)row * OUT_DIM + lm;
            zr[0]  = c0[r];
            zr[16] = c1[r];
            zr[32] = c2[r];
            zr[48] = c3[r];
        }
    }
}

// ---------------- per-node attention scalars s,d (one wave per node) --------
__global__ void k_attn(const float* __restrict__ z, const float* __restrict__ attn,
                       float* __restrict__ s, float* __restrict__ d, int n) {
    const int lane = threadIdx.x & 31;
    const int node = blockIdx.x * (blockDim.x >> 5) + (threadIdx.x >> 5);
    if (node >= n) return;
    const float* zr = z + (size_t)node * OUT_DIM;
    float z0 = zr[lane], z1 = zr[lane + 32];
    float ss = z0 * attn[lane]      + z1 * attn[lane + 32];
    float dd = z0 * attn[64 + lane] + z1 * attn[96 + lane];
    for (int off = 16; off; off >>= 1) {
        ss += __shfl_xor(ss, off, 32);
        dd += __shfl_xor(dd, off, 32);
    }
    if (lane == 0) { s[node] = ss; d[node] = dd; }
}

// ---------------- zero deg + cursor -----------------------------------------
__global__ void k_zero(int* __restrict__ deg, int* __restrict__ cur, int n) {
    int i = blockIdx.x * blockDim.x + threadIdx.x;
    if (i < n) { deg[i] = 0; cur[i] = 0; }
}

// ---------------- degree histogram ------------------------------------------
__global__ void k_deg(const int* __restrict__ dst, int* __restrict__ deg, int E) {
    int e = blockIdx.x * blockDim.x + threadIdx.x;
    if (e < E) atomicAdd(&deg[dst[e]], 1);
}

// ---------------- single-workgroup exclusive scan (CSR offsets) -------------
#define SCAN_T 1024
__global__ void k_scan(const int* __restrict__ deg, int* __restrict__ off, int n) {
    __shared__ int tile[SCAN_T];
    __shared__ int carry_s;
    const int t = threadIdx.x;
    if (t == 0) carry_s = 0;
    __syncthreads();
    for (int base = 0; base < n; base += SCAN_T) {
        int i = base + t;
        int v = (i < n) ? deg[i] : 0;
        tile[t] = v;
        __syncthreads();
        for (int o = 1; o < SCAN_T; o <<= 1) {
            int add = (t >= o) ? tile[t - o] : 0;
            __syncthreads();
            tile[t] += add;
            __syncthreads();
        }
        int incl = tile[t];
        int c = carry_s;
        if (i < n) off[i] = c + incl - v;   // exclusive
        __syncthreads();
        if (t == SCAN_T - 1) carry_s = c + incl;
        __syncthreads();
    }
    if (t == 0) off[n] = carry_s;
}

// ---------------- scatter src ids into CSR order ----------------------------
__global__ void k_sort(const int* __restrict__ src, const int* __restrict__ dst,
                       const int* __restrict__ off, int* __restrict__ cur,
                       int* __restrict__ ssrc, int E) {
    int e = blockIdx.x * blockDim.x + threadIdx.x;
    if (e >= E) return;
    int di = dst[e];
    int p = off[di] + atomicAdd(&cur[di], 1);
    ssrc[p] = src[e];
}

// ---------------- per-node softmax aggregation (one wave per node) ----------
__global__ void k_node(const int* __restrict__ off, const int* __restrict__ ssrc,
                       const float* __restrict__ s, const float* __restrict__ dv,
                       const float* __restrict__ z, float* __restrict__ out, int n) {
    const int lane = threadIdx.x & 31;
    const int node = blockIdx.x * (blockDim.x >> 5) + (threadIdx.x >> 5);
    if (node >= n) return;

    const int beg = off[node];
    const int end = off[node + 1];
    const float di = dv[node];

    // pass 1: segment max (lane-parallel + shuffle reduce)
    float mx = -3.4e38f;
    for (int j = beg + lane; j < end; j += 32) {
        float ev = s[ssrc[j]] + di;
        ev = ev > 0.0f ? ev : ev * NEG_SLOPE;
        mx = fmaxf(mx, ev);
    }
    for (int o = 16; o; o >>= 1)
        mx = fmaxf(mx, __shfl_xor(mx, o, 32));
    const float m = (end > beg) ? mx : 0.0f;   // isfinite(m) ? m : 0

    // pass 2: den + num (sequential over edges; w identical across lanes,
    //         each lane owns columns lane and lane+32 -> no atomics, no reduce)
    float den = 0.0f, acc0 = 0.0f, acc1 = 0.0f;
    for (int j = beg; j < end; ++j) {
        int sj = ssrc[j];                      // broadcast load
        float ev = s[sj] + di;
        ev = ev > 0.0f ? ev : ev * NEG_SLOPE;
        float w = __expf(ev - m);
        den += w;
        const float* zs = z + (size_t)sj * OUT_DIM;
        acc0 += w * zs[lane];
        acc1 += w * zs[lane + 32];
    }

    float* op = out + (size_t)node * OUT_DIM;
    if (den > 0.0f) {
        float inv = 1.0f / fmaxf(den, 1e-30f);
        op[lane]      = acc0 * inv;
        op[lane + 32] = acc1 * inv;
    } else {
        op[lane]      = 0.0f;
        op[lane + 32] = 0.0f;
    }
}

extern "C" void kernel_launch(void* const* d_in, const int* in_sizes, int n_in,
                              void* d_out, int out_size, void* d_ws, size_t ws_size,
                              hipStream_t stream) {
    const float* h    = (const float*)d_in[0];
    const int*   src  = (const int*)d_in[1];
    const int*   dst  = (const int*)d_in[2];
    const float* fcw  = (const float*)d_in[3];
    const float* attn = (const float*)d_in[4];
    float* out = (float*)d_out;

    const int N = in_sizes[0] / IN_DIM;   // 100000
    const int E = in_sizes[1];            // 1600000

    // workspace layout (~34.0 MB)
    float* z    = (float*)d_ws;                    // N*64
    float* s    = z + (size_t)N * OUT_DIM;         // N
    float* dv   = s + N;                           // N
    int*   deg  = (int*)(dv + N);                  // N
    int*   off  = deg + N;                         // N+1
    int*   cur  = off + N + 1;                     // N
    int*   ssrc = cur + N;                         // E

    const int B = 256;
    k_gemm<<<(N + 127) / 128, B, 0, stream>>>(h, fcw, z, N);
    k_attn<<<(N + 7) / 8, B, 0, stream>>>(z, attn, s, dv, N);
    k_zero<<<(N + B - 1) / B, B, 0, stream>>>(deg, cur, N);
    k_deg<<<(E + B - 1) / B, B, 0, stream>>>(dst, deg, E);
    k_scan<<<1, SCAN_T, 0, stream>>>(deg, off, N);
    k_sort<<<(E + B - 1) / B, B, 0, stream>>>(src, dst, off, cur, ssrc, E);
    k_node<<<(N + 7) / 8, B, 0, stream>>>(off, ssrc, s, dv, z, out, N);
}